// GraphSAGE_8787503088149
// MI455X (gfx1250) — compile-verified
//
#include <hip/hip_runtime.h>
#include <hip/hip_bf16.h>

typedef __attribute__((ext_vector_type(2))) float v2f;
typedef __attribute__((ext_vector_type(8))) float v8f;

#define D_FEAT 96
#define KSTEPS 24        // 96 / 4
#define N_LAYERS 3
#define BN_EPS 1e-5f

// ---------------------------------------------------------------- utilities
__global__ void zero_f32_kernel(float* __restrict__ p, int n) {
    int i = blockIdx.x * blockDim.x + threadIdx.x;
    if (i < n) p[i] = 0.0f;
}

__global__ void degree_kernel(const int* __restrict__ dst, float* __restrict__ deg, int n_edges) {
    int e = blockIdx.x * blockDim.x + threadIdx.x;
    if (e < n_edges) atomicAdd(&deg[dst[e]], 1.0f);
}

__global__ void invdeg_kernel(float* __restrict__ deg, int n) {
    int i = blockIdx.x * blockDim.x + threadIdx.x;
    if (i < n) {
        float d = deg[i];
        deg[i] = 1.0f / fmaxf(d, 1.0f);
    }
}

// ------------------------------------------------- edge gather / scatter-add
// 12 threads per edge, 8 consecutive features per thread.
__global__ void aggregate_kernel(const float* __restrict__ h_in,
                                 const int* __restrict__ src,
                                 const int* __restrict__ dst,
                                 float* __restrict__ agg, int n_edges) {
    int tid = blockIdx.x * blockDim.x + threadIdx.x;
    int e = tid / 12;
    int f = (tid - e * 12) * 8;
    if (e >= n_edges) return;
    int s = src[e];
    int d = dst[e];
    const float* hs = h_in + (size_t)s * D_FEAT + f;
    float* ad = agg + (size_t)d * D_FEAT + f;
#pragma unroll
    for (int i = 0; i < 8; ++i) atomicAdd(&ad[i], hs[i]);
}

// ---------------------------------------------------------- fused SAGE GEMM
// h_pre = (agg * inv_deg) @ Wn + h_in @ Ws + bias
// Also accumulates per-column sum / sumsq into stats[0..95] / stats[96..191].
// grid = (blocksX, 6); blockIdx.y selects the 16-column tile; one wave per
// 16x16 output tile, grid-stride over row tiles.
__global__ void __launch_bounds__(256)
sage_gemm_kernel(const float* __restrict__ agg, const float* __restrict__ h_in,
                 const float* __restrict__ inv_deg,
                 const float* __restrict__ Wn, const float* __restrict__ Ws,
                 const float* __restrict__ bias,
                 float* __restrict__ h_pre, float* __restrict__ stats,
                 int n_nodes) {
    const int col0 = blockIdx.y * 16;
    __shared__ float sWn[D_FEAT * 16];
    __shared__ float sWs[D_FEAT * 16];
    for (int i = threadIdx.x; i < D_FEAT * 16; i += blockDim.x) {
        int k = i >> 4, c = i & 15;
        sWn[i] = Wn[k * D_FEAT + col0 + c];
        sWs[i] = Ws[k * D_FEAT + col0 + c];
    }
    __syncthreads();

    const int lane = threadIdx.x & 31;
    const int half = lane >> 4;      // 0: K pair {0,1}   1: K pair {2,3}
    const int l16 = lane & 15;
    const int waveId = blockIdx.x * (blockDim.x >> 5) + (threadIdx.x >> 5);
    const int nWaves = gridDim.x * (blockDim.x >> 5);
    const int mTiles = n_nodes >> 4;

    for (int tm = waveId; tm < mTiles; tm += nWaves) {
        const int row0 = tm * 16;
        const int arow = row0 + l16;                 // A-matrix row for this lane
        const float idg = inv_deg[arow];
        const float* aggRow = agg + (size_t)arow * D_FEAT;
        const float* hRow = h_in + (size_t)arow * D_FEAT;

        v8f acc = {};
        for (int ks = 0; ks < KSTEPS; ++ks) {
            const int k = ks * 4 + half * 2;
            v2f a, b;
            a.x = aggRow[k] * idg;
            a.y = aggRow[k + 1] * idg;
            b.x = sWn[k * 16 + l16];
            b.y = sWn[(k + 1) * 16 + l16];
            acc = __builtin_amdgcn_wmma_f32_16x16x4_f32(false, a, false, b,
                                                        (short)0, acc, false, false);
            v2f a2, b2;
            a2.x = hRow[k];
            a2.y = hRow[k + 1];
            b2.x = sWs[k * 16 + l16];
            b2.y = sWs[(k + 1) * 16 + l16];
            acc = __builtin_amdgcn_wmma_f32_16x16x4_f32(false, a2, false, b2,
                                                        (short)0, acc, false, false);
        }

        const int col = col0 + l16;                  // C/D column for this lane
        const float bv = bias[col];
        const int rbase = row0 + half * 8;           // C/D rows: half0 -> 0..7, half1 -> 8..15
        float s = 0.0f, sq = 0.0f;
#pragma unroll
        for (int r = 0; r < 8; ++r) {
            float v = acc[r] + bv;
            h_pre[(size_t)(rbase + r) * D_FEAT + col] = v;
            s += v;
            sq += v * v;
        }
        atomicAdd(&stats[col], s);
        atomicAdd(&stats[D_FEAT + col], sq);
    }
}

// ------------------------------------------------------------- batchnorm
__global__ void bn_finalize_kernel(const float* __restrict__ stats,
                                   const float* __restrict__ gamma,
                                   const float* __restrict__ beta,
                                   float* __restrict__ scale,
                                   float* __restrict__ shift, int n_nodes) {
    int c = threadIdx.x;
    if (c < D_FEAT) {
        float inv_n = 1.0f / (float)n_nodes;
        float mu = stats[c] * inv_n;
        float var = stats[D_FEAT + c] * inv_n - mu * mu;
        float rs = rsqrtf(var + BN_EPS);
        float sc = gamma[c] * rs;
        scale[c] = sc;
        shift[c] = beta[c] - mu * sc;
    }
}

// normalize + ReLU, and zero `agg` for the next layer in the same pass
__global__ void bn_apply_kernel(const float* __restrict__ h_pre,
                                const float* __restrict__ scale,
                                const float* __restrict__ shift,
                                float* __restrict__ h_out,
                                float* __restrict__ agg, int total) {
    int i = blockIdx.x * blockDim.x + threadIdx.x;
    if (i < total) {
        int c = i % D_FEAT;
        float v = fmaf(h_pre[i], scale[c], shift[c]);
        h_out[i] = fmaxf(v, 0.0f);
        agg[i] = 0.0f;
    }
}

// ---------------------------------------------------------- classifier GEMM
// out[N,10] = h @ clf_w[96,10] + clf_b ; B padded to 96x16 in LDS.
__global__ void __launch_bounds__(256)
clf_kernel(const float* __restrict__ h, const float* __restrict__ W,
           const float* __restrict__ b, float* __restrict__ out, int n_nodes) {
    __shared__ float sW[D_FEAT * 16];
    for (int i = threadIdx.x; i < D_FEAT * 16; i += blockDim.x) {
        int k = i >> 4, c = i & 15;
        sW[i] = (c < 10) ? W[k * 10 + c] : 0.0f;
    }
    __syncthreads();

    const int lane = threadIdx.x & 31;
    const int half = lane >> 4;
    const int l16 = lane & 15;
    const int waveId = blockIdx.x * (blockDim.x >> 5) + (threadIdx.x >> 5);
    const int nWaves = gridDim.x * (blockDim.x >> 5);
    const int mTiles = n_nodes >> 4;

    for (int tm = waveId; tm < mTiles; tm += nWaves) {
        const int row0 = tm * 16;
        const int arow = row0 + l16;
        const float* hRow = h + (size_t)arow * D_FEAT;

        v8f acc = {};
        for (int ks = 0; ks < KSTEPS; ++ks) {
            const int k = ks * 4 + half * 2;
            v2f a, bb;
            a.x = hRow[k];
            a.y = hRow[k + 1];
            bb.x = sW[k * 16 + l16];
            bb.y = sW[(k + 1) * 16 + l16];
            acc = __builtin_amdgcn_wmma_f32_16x16x4_f32(false, a, false, bb,
                                                        (short)0, acc, false, false);
        }

        const int col = l16;
        if (col < 10) {
            const float bv = b[col];
            const int rbase = row0 + half * 8;
#pragma unroll
            for (int r = 0; r < 8; ++r)
                out[(size_t)(rbase + r) * 10 + col] = acc[r] + bv;
        }
    }
}

// ------------------------------------------------------------------- launch
extern "C" void kernel_launch(void* const* d_in, const int* in_sizes, int n_in,
                              void* d_out, int out_size, void* d_ws, size_t ws_size,
                              hipStream_t stream) {
    const float* x      = (const float*)d_in[0];
    const int*   edge   = (const int*)d_in[1];
    const float* w_nei  = (const float*)d_in[2];   // [3,96,96]
    const float* w_self = (const float*)d_in[3];   // [3,96,96]
    const float* bias   = (const float*)d_in[4];   // [3,96]
    const float* gamma  = (const float*)d_in[5];   // [3,96]
    const float* beta   = (const float*)d_in[6];   // [3,96]
    const float* clf_w  = (const float*)d_in[7];   // [96,10]
    const float* clf_b  = (const float*)d_in[8];   // [10]
    float* out = (float*)d_out;

    const int N = in_sizes[0] / D_FEAT;   // 50000
    const int E = in_sizes[1] / 2;        // 800000
    const int* src = edge;
    const int* dst = edge + E;

    // workspace layout (all f32)
    float* ws = (float*)d_ws;
    float* inv_deg = ws;                         // N   (deg, then inverted in place)
    float* h_buf   = inv_deg + N;                // N*96
    float* h_pre   = h_buf + (size_t)N * D_FEAT; // N*96
    float* agg     = h_pre + (size_t)N * D_FEAT; // N*96
    float* stats   = agg + (size_t)N * D_FEAT;   // 192 (sum | sumsq)
    float* scale   = stats + 192;                // 96
    float* shift   = scale + D_FEAT;             // 96

    const int T = N * D_FEAT;
    const int B256 = 256;

    // init: deg = 0, agg = 0, stats = 0
    zero_f32_kernel<<<(N + B256 - 1) / B256, B256, 0, stream>>>(inv_deg, N);
    zero_f32_kernel<<<(T + B256 - 1) / B256, B256, 0, stream>>>(agg, T);
    zero_f32_kernel<<<1, B256, 0, stream>>>(stats, 192);

    degree_kernel<<<(E + B256 - 1) / B256, B256, 0, stream>>>(dst, inv_deg, E);
    invdeg_kernel<<<(N + B256 - 1) / B256, B256, 0, stream>>>(inv_deg, N);

    for (int layer = 0; layer < N_LAYERS; ++layer) {
        const float* h_in = (layer == 0) ? x : h_buf;
        const float* Wn = w_nei + (size_t)layer * D_FEAT * D_FEAT;
        const float* Wsf = w_self + (size_t)layer * D_FEAT * D_FEAT;

        long long aggThreads = (long long)E * 12;
        aggregate_kernel<<<(int)((aggThreads + B256 - 1) / B256), B256, 0, stream>>>(
            h_in, src, dst, agg, E);

        dim3 gemmGrid(64, D_FEAT / 16);   // 64 blocks x 6 column tiles, 8 waves/block
        sage_gemm_kernel<<<gemmGrid, B256, 0, stream>>>(
            agg, h_in, inv_deg, Wn, Wsf, bias + layer * D_FEAT,
            h_pre, stats, N);

        bn_finalize_kernel<<<1, 128, 0, stream>>>(
            stats, gamma + layer * D_FEAT, beta + layer * D_FEAT, scale, shift, N);
        zero_f32_kernel<<<1, B256, 0, stream>>>(stats, 192);

        bn_apply_kernel<<<(T + B256 - 1) / B256, B256, 0, stream>>>(
            h_pre, scale, shift, h_buf, agg, T);
    }

    clf_kernel<<<128, B256, 0, stream>>>(h_buf, clf_w, clf_b, out, N);
}